// TopPCrossEntropy_12799002542299
// MI455X (gfx1250) — compile-verified
//
#include <hip/hip_runtime.h>
#include <hip/hip_bf16.h>
#include <math.h>

#define NCLS 32000
#define BLK  1024
#define TOPP 0.9f

// ---------------- CDNA5 async global->LDS helpers (gfx1250) ----------------
typedef int v4i __attribute__((vector_size(16)));
typedef __attribute__((address_space(1))) v4i* g_v4p;   // global int4*
typedef __attribute__((address_space(3))) v4i* l_v4p;   // LDS int4*

__device__ __forceinline__ void async_g2l_b128(const void* g, void* l) {
#if __has_builtin(__builtin_amdgcn_global_load_async_to_lds_b128)
  __builtin_amdgcn_global_load_async_to_lds_b128(
      (g_v4p)(unsigned long long)g,
      (l_v4p)(unsigned)(unsigned long long)l, 0, 0);
#else
  asm volatile("global_load_async_to_lds_b128 %0, %1, off"
               :
               : "v"((unsigned)(unsigned long long)l), "v"(g)
               : "memory");
#endif
}

__device__ __forceinline__ void wait_async0() {
#if __has_builtin(__builtin_amdgcn_s_wait_asynccnt)
  __builtin_amdgcn_s_wait_asynccnt(0);
#else
  asm volatile("s_wait_asynccnt 0" ::: "memory");
#endif
}

// ---------------- wave32 / block reductions ----------------
__device__ __forceinline__ float wave_sum(float v) {
  for (int m = 16; m >= 1; m >>= 1) v += __shfl_xor(v, m, 32);
  return v;
}
__device__ __forceinline__ float wave_max(float v) {
  for (int m = 16; m >= 1; m >>= 1) v = fmaxf(v, __shfl_xor(v, m, 32));
  return v;
}

__device__ float block_sum(float v, float* red) {
  const int tid = threadIdx.x, lane = tid & 31, wave = tid >> 5;
  v = wave_sum(v);
  if (lane == 0) red[wave] = v;
  __syncthreads();
  if (wave == 0) {
    float x = (lane < (int)(blockDim.x >> 5)) ? red[lane] : 0.0f;
    x = wave_sum(x);
    if (lane == 0) red[0] = x;
  }
  __syncthreads();
  const float r = red[0];
  __syncthreads();
  return r;
}

__device__ float block_max(float v, float* red) {
  const int tid = threadIdx.x, lane = tid & 31, wave = tid >> 5;
  v = wave_max(v);
  if (lane == 0) red[wave] = v;
  __syncthreads();
  if (wave == 0) {
    float x = (lane < (int)(blockDim.x >> 5)) ? red[lane] : -INFINITY;
    x = wave_max(x);
    if (lane == 0) red[0] = x;
  }
  __syncthreads();
  const float r = red[0];
  __syncthreads();
  return r;
}

// ---------------- per-row top-p cross-entropy ----------------
// One workgroup (32 wave32s) per row. Both rows staged once into LDS via
// async b128 global->LDS copies; all later passes are LDS-resident.
__global__ __launch_bounds__(BLK) void toppce_rows(const float* __restrict__ student,
                                                   const float* __restrict__ teacher,
                                                   float* __restrict__ row_ce) {
  extern __shared__ float smem[];
  float*    sl    = smem;                                  // [NCLS] raw student logits
  float*    tu    = smem + NCLS;                           // [NCLS] teacher -> u = exp(t - tmax)
  float*    hmass = smem + 2 * NCLS;                       // [256] bucket mass
  unsigned* hcnt  = (unsigned*)(smem + 2 * NCLS + 256);    // [256] bucket count
  float*    red   = smem + 2 * NCLS + 512;                 // [64]  reduction scratch
  float*    scf   = smem + 2 * NCLS + 576;                 // [8]   scalar floats
  unsigned* scu   = (unsigned*)(smem + 2 * NCLS + 584);    // [8]   scalar uints

  const int tid = threadIdx.x;
  const long long row = blockIdx.x;
  const float* srow = student + row * (long long)NCLS;
  const float* trow = teacher + row * (long long)NCLS;

  // Stage both rows: single HBM read each (ASYNCcnt-tracked b128 copies).
  for (int i = tid * 4; i < NCLS; i += BLK * 4) {
    async_g2l_b128(trow + i, tu + i);
    async_g2l_b128(srow + i, sl + i);
  }
  wait_async0();
  __syncthreads();

  // Pass A: row maxima (teacher & student).
  float tmx = -INFINITY, smx = -INFINITY;
  for (int i = tid; i < NCLS; i += BLK) {
    tmx = fmaxf(tmx, tu[i]);
    smx = fmaxf(smx, sl[i]);
  }
  tmx = block_max(tmx, red);
  smx = block_max(smx, red);

  // Pass B: u = exp(t - tmx) written back; Zt, Zs.
  float zt = 0.f, zs = 0.f;
  for (int i = tid; i < NCLS; i += BLK) {
    const float u = __expf(tu[i] - tmx);
    tu[i] = u;
    zt += u;
    zs += __expf(sl[i] - smx);
  }
  zt = block_sum(zt, red);
  zs = block_sum(zs, red);
  const float target = TOPP * zt;  // top-p boundary in unnormalized mass

  // Radix select (4 x 8 bits on float bit pattern of u, all positive):
  // find boundary so kept set = {u : bits(u) > vbits} (or >= for edge cases).
  if (tid == 0) { scf[0] = 0.f; scu[0] = 0u; scu[1] = 0u; scu[2] = 0u; }
  __syncthreads();

  for (int round = 0; round < 4; ++round) {
    const int shift = 24 - 8 * round;
    if (tid < 256) { hmass[tid] = 0.f; hcnt[tid] = 0u; }
    __syncthreads();
    const unsigned state = scu[2];
    const unsigned pval  = scu[0];
    const unsigned pmask = (round == 0) ? 0u : (0xFFFFFFFFu << (shift + 8));
    if (state == 0u) {
      for (int i = tid; i < NCLS; i += BLK) {
        const float u = tu[i];
        const unsigned bits = __float_as_uint(u);
        if ((bits & pmask) == pval) {
          const unsigned b = (bits >> shift) & 0xFFu;
          atomicAdd(&hmass[b], u);   // ds_add_f32
          atomicAdd(&hcnt[b], 1u);   // ds_add_u32
        }
      }
    }
    __syncthreads();
    if (tid == 0 && state == 0u) {
      float acc = scf[0];
      unsigned cacc = scu[1];
      int chosen = -1; float cm = 0.f;
      for (int b = 255; b >= 0; --b) {
        const unsigned c = hcnt[b];
        if (c == 0u) continue;
        const float m = hmass[b];
        if (acc + m <= target) { acc += m; cacc += c; }
        else { chosen = b; cm = m; break; }
      }
      if (chosen < 0) {
        // everything at/below current prefix is kept (inclusive threshold)
        scu[2] = 1u; scu[3] = 1u; scu[4] = pval; scf[2] = acc;
      } else {
        const unsigned npref = pval | (((unsigned)chosen) << shift);
        scu[0] = npref; scf[0] = acc; scu[1] = cacc;
        if (round == 3) {
          scu[2] = 1u;
          if (cacc == 0u) { scu[3] = 1u; scu[4] = npref; scf[2] = cm; }   // force top-1
          else            { scu[3] = 0u; scu[4] = npref; scf[2] = acc; }  // strict >
        }
      }
    }
    __syncthreads();
  }

  const unsigned ge    = scu[3];
  const unsigned vbits = scu[4];
  const float    mass  = scf[2];

  // Pass C: masked dot of kept u with raw student logits.
  // CE_row = -(sum_kept u*s)/mass + smax + log(Zs)   (since sum_kept u = mass)
  float dot = 0.f;
  for (int i = tid; i < NCLS; i += BLK) {
    const unsigned bits = __float_as_uint(tu[i]);
    const bool keep = ge ? (bits >= vbits) : (bits > vbits);
    if (keep) dot += tu[i] * sl[i];
  }
  dot = block_sum(dot, red);

  if (tid == 0) {
    row_ce[row] = -(dot / mass) + (smx + __logf(zs));
  }
}

// Deterministic mean over rows -> scalar output.
__global__ __launch_bounds__(BLK) void mean_rows(const float* __restrict__ row_ce,
                                                 float* __restrict__ out, int n) {
  __shared__ float red[64];
  float s = 0.f;
  for (int i = threadIdx.x; i < n; i += BLK) s += row_ce[i];
  s = block_sum(s, red);
  if (threadIdx.x == 0) out[0] = s / (float)n;
}

extern "C" void kernel_launch(void* const* d_in, const int* in_sizes, int n_in,
                              void* d_out, int out_size, void* d_ws, size_t ws_size,
                              hipStream_t stream) {
  const float* student = (const float*)d_in[0];
  const float* teacher = (const float*)d_in[1];
  float* out    = (float*)d_out;
  float* row_ce = (float*)d_ws;  // batch floats of scratch

  const int batch = in_sizes[0] / NCLS;
  const size_t shmem = (size_t)(2 * NCLS + 512 + 64 + 16) * sizeof(float);  // 258368 B

  (void)hipFuncSetAttribute((const void*)toppce_rows,
                            hipFuncAttributeMaxDynamicSharedMemorySize, (int)shmem);

  toppce_rows<<<batch, BLK, shmem, stream>>>(student, teacher, row_ce);
  mean_rows<<<1, BLK, 0, stream>>>(row_ce, out, batch);
}